// LinearAutoregressiveHMM_55284819034907
// MI455X (gfx1250) — compile-verified
//
#include <hip/hip_runtime.h>
#include <math.h>

typedef float v2f __attribute__((ext_vector_type(2)));
typedef float v8f __attribute__((ext_vector_type(8)));

#define L_ 256
#define LOG2PI_F 1.8378770664093453f
#define NKK 9   // 9 k-steps of 4: K = 32 data dims + 4 augmented (1, 0, 0, 0)

// workspace float offsets
#define WS_BFRAG 0                         // 32 states * 2 ntile * 9 kstep * 32 lane * float2
#define WS_K     (32 * 2 * NKK * 32 * 2)   // 32  : logdet + C*log(2pi) per state
#define WS_LOGT  (WS_K + 32)               // 32*32
#define WS_TRANS (WS_LOGT + 1024)          // 32*32
#define WS_LI    (WS_TRANS + 1024)         // 32

// DPP-based add of lane (lane ^ pattern) value; CTRL is a DPP16 control code.
template <int CTRL>
__device__ __forceinline__ float dpp_add(float v) {
  int x = __builtin_amdgcn_update_dpp(0, __float_as_int(v), CTRL, 0xF, 0xF, true);
  return v + __int_as_float(x);
}

__device__ __forceinline__ float bcast_lane(float v, int lane) {
  return __int_as_float(__builtin_amdgcn_readlane(__float_as_int(v), lane));
}

// ---------------------------------------------------------------------------
// Prep: per-state Cholesky -> triangular inverse -> WMMA-B-fragment packing
// (with affine augmentation row -w), plus softmax tables. One wave, thread s
// owns state s. Tiny one-off cost.
// ---------------------------------------------------------------------------
__global__ __launch_bounds__(32) void hmm_prep(
    const float* __restrict__ trans_mat,
    const float* __restrict__ init_dist,
    const float* __restrict__ means,
    const float* __restrict__ cc,
    float* __restrict__ ws)
{
  const int s = threadIdx.x; // 0..31

  float T[32][32];   // scratch: tril, later Linv
  float A[32][32];   // scratch: covar, later its Cholesky factor

  // tril with exp(diag)
  for (int i = 0; i < 32; i++)
    for (int j = 0; j < 32; j++) {
      float v = cc[(s * 32 + i) * 32 + j];
      T[i][j] = (j < i) ? v : ((j == i) ? __expf(v) : 0.0f);
    }
  // covar = T T^T + 1e-6 I (lower triangle)
  for (int i = 0; i < 32; i++)
    for (int j = 0; j <= i; j++) {
      float acc = 0.0f;
      for (int k = 0; k <= j; k++) acc += T[i][k] * T[j][k];
      if (i == j) acc += 1e-6f;
      A[i][j] = acc;
    }
  // Cholesky in place (lower)
  float logdet = 0.0f;
  for (int j = 0; j < 32; j++) {
    float d = A[j][j];
    for (int k = 0; k < j; k++) d -= A[j][k] * A[j][k];
    d = sqrtf(d);
    A[j][j] = d;
    logdet += __logf(d);
    float inv = 1.0f / d;
    for (int i = j + 1; i < 32; i++) {
      float acc = A[i][j];
      for (int k = 0; k < j; k++) acc -= A[i][k] * A[j][k];
      A[i][j] = acc * inv;
    }
  }
  logdet *= 2.0f;

  // Linv = inv(chol), lower triangular, into T
  for (int i = 0; i < 32; i++)
    for (int j = 0; j < 32; j++) T[i][j] = 0.0f;
  for (int j = 0; j < 32; j++) {
    T[j][j] = 1.0f / A[j][j];
    for (int i = j + 1; i < 32; i++) {
      float acc = 0.0f;
      for (int k = j; k < i; k++) acc += A[i][k] * T[k][j];
      T[i][j] = -acc / A[i][i];
    }
  }

  // w = Linv @ mu ; per-state constant = logdet + C*log(2pi)
  float w[32];
  for (int i = 0; i < 32; i++) {
    float acc = 0.0f;
    for (int j = 0; j <= i; j++) acc += T[i][j] * means[s * 32 + j];
    w[i] = acc;
  }
  ws[WS_K + s] = logdet + 32.0f * LOG2PI_F;

  // Pack Linv^T into V_WMMA_F32_16X16X4_F32 B-fragment order, K = 36:
  //   k < 32 : B[k][n] = Linv[nCol][k]          (Y = X @ Linv^T)
  //   k = 32 : B[32][n] = -w[nCol], k = 33..35 : 0   (affine: Y = Linv x - w)
  // lane<16: (K=4kk+0,+1, N=lane) ; lane>=16: (K=4kk+2,+3, N=lane-16).
  v2f* bf = (v2f*)ws;
  for (int n = 0; n < 2; n++) {
    for (int kk = 0; kk < 8; kk++)
      for (int lane = 0; lane < 32; lane++) {
        int k0  = 4 * kk + ((lane < 16) ? 0 : 2);
        int col = n * 16 + (lane & 15);
        v2f v;
        v.x = T[col][k0];
        v.y = T[col][k0 + 1];
        bf[((s * 2 + n) * NKK + kk) * 32 + lane] = v;
      }
    for (int lane = 0; lane < 32; lane++) {   // kk = 8 (augmented row)
      int col = n * 16 + (lane & 15);
      v2f v;
      v.x = (lane < 16) ? -w[col] : 0.0f;     // K=32 row is -w; K=33..35 zero
      v.y = 0.0f;
      bf[((s * 2 + n) * NKK + 8) * 32 + lane] = v;
    }
  }

  // transition softmax row s -> trans, log(trans+1e-8)
  {
    float x[32];
    float m = -INFINITY;
    for (int j = 0; j < 32; j++) { x[j] = trans_mat[s * 32 + j]; m = fmaxf(m, x[j]); }
    float sum = 0.0f;
    for (int j = 0; j < 32; j++) { x[j] = __expf(x[j] - m); sum += x[j]; }
    float r = 1.0f / sum;
    for (int j = 0; j < 32; j++) {
      float p = x[j] * r;
      ws[WS_TRANS + s * 32 + j] = p;
      ws[WS_LOGT + s * 32 + j]  = __logf(p + 1e-8f);
    }
  }
  // log softmax of initial distribution
  {
    float m = -INFINITY;
    for (int j = 0; j < 32; j++) m = fmaxf(m, init_dist[j]);
    float sum = 0.0f;
    for (int j = 0; j < 32; j++) sum += __expf(init_dist[j] - m);
    ws[WS_LI + s] = init_dist[s] - (m + __logf(sum));
  }
}

// ---------------------------------------------------------------------------
// Main fused kernel: one block (8 wave32) per batch element.
//  B: affine mahalanobis via V_WMMA_F32_16X16X4_F32 (K=36), square + 4 DPP
//     adds, store -0.5*mahal to LDS. No per-state LDS reads in the hot loop.
//  C: Viterbi (wave 0, lane = state, v_readlane broadcast; per-state constant
//     folded in here as a register add)
//  D: AR dot + means[next_state]
// ---------------------------------------------------------------------------
__global__ __launch_bounds__(256) void hmm_main(
    const float* __restrict__ E,     // (256,256,32)
    const float* __restrict__ Wm,    // (256,32,1)
    const float* __restrict__ means, // (32,32)
    const float* __restrict__ ws,
    float* __restrict__ out)         // (256,1,32)
{
  __shared__ float s_emlp[L_][32];
  __shared__ float s_logT[32][32];
  __shared__ float s_trans[32][32];
  __shared__ float s_k[32];
  __shared__ float s_li[32];
  __shared__ float s_red[8][32];
  __shared__ int   s_state;

  const int b    = blockIdx.x;
  const int tid  = threadIdx.x;
  const int wave = tid >> 5;
  const int lane = tid & 31;

  // ---- Phase A: stage tables -------------------------------------------
  for (int i = tid; i < 1024; i += 256) {
    s_logT[i >> 5][i & 31]  = ws[WS_LOGT + i];
    s_trans[i >> 5][i & 31] = ws[WS_TRANS + i];
  }
  if (tid < 32) { s_k[tid] = ws[WS_K + tid]; s_li[tid] = ws[WS_LI + tid]; }
  __syncthreads();

  // ---- Phase B: emission -0.5*mahal via fp32 WMMA (affine, K=36) --------
  const float* Eb = E + (size_t)b * L_ * 32;
  const v2f*   bf = (const v2f*)ws;
  const int halfsel = (lane < 16) ? 0 : 2; // K sub-offset per A-layout
  const int lcol    = lane & 15;

  // augmented A fragment: K=32 element is the constant 1 (lanes<16), rest 0
  v2f aug;
  aug.x = (lane < 16) ? 1.0f : 0.0f;
  aug.y = 0.0f;

  for (int rt = wave; rt < 16; rt += 8) {      // 16 row-tiles of 16 timesteps
    const int t0 = rt * 16;
    v2f a[NKK];
    const float* arow = Eb + (t0 + lcol) * 32 + halfsel;
#pragma unroll
    for (int kk = 0; kk < 8; kk++)
      a[kk] = *(const v2f*)(arow + 4 * kk);
    a[8] = aug;

    for (int s = 0; s < 32; s++) {
      v8f acc0 = {0.f, 0.f, 0.f, 0.f, 0.f, 0.f, 0.f, 0.f};
      v8f acc1 = {0.f, 0.f, 0.f, 0.f, 0.f, 0.f, 0.f, 0.f};
      const v2f* bfs = bf + (size_t)(s * 2) * NKK * 32;
#pragma unroll
      for (int kk = 0; kk < NKK; kk++) {
        v2f b0 = bfs[kk * 32 + lane];
        v2f b1 = bfs[(NKK + kk) * 32 + lane];
        acc0 = __builtin_amdgcn_wmma_f32_16x16x4_f32(
                   false, a[kk], false, b0, (short)0, acc0, false, false);
        acc1 = __builtin_amdgcn_wmma_f32_16x16x4_f32(
                   false, a[kk], false, b1, (short)0, acc1, false, false);
      }
#pragma unroll
      for (int k = 0; k < 8; k++) {
        float y0 = acc0[k], y1 = acc1[k];
        float p = y0 * y0 + y1 * y1;       // (Linv(x-mu))^2 partial
        // 16-lane column sum entirely in the VALU via DPP:
        p = dpp_add<0x0B1>(p);  // xor 1 : quad_perm(1,0,3,2)
        p = dpp_add<0x04E>(p);  // xor 2 : quad_perm(2,3,0,1)
        p = dpp_add<0x141>(p);  // xor 4 : row_half_mirror (quads uniform)
        p = dpp_add<0x140>(p);  // xor 8 : row_mirror (8-groups uniform)
        if (lcol == 0) {
          int row = t0 + k + ((lane >> 4) << 3); // lanes>=16 hold M=8..15
          s_emlp[row][s] = -0.5f * p;            // constant added in Viterbi
        }
      }
    }
  }
  __syncthreads();

  // ---- Phase C: Viterbi (wave 0, lane j = state j) ----------------------
  if (wave == 0) {
    float lt[32];
#pragma unroll
    for (int i = 0; i < 32; i++) lt[i] = s_logT[i][lane]; // column j
    const float kadj = -0.5f * s_k[lane]; // -(logdet + C*log2pi)/2 for state j
    float delta = s_li[lane] + s_emlp[0][lane] + kadj;
    for (int t = 1; t < L_; t++) {
      float em = s_emlp[t][lane] + kadj;
      float m = -INFINITY;
#pragma unroll
      for (int i = 0; i < 32; i++) {
        float di = bcast_lane(delta, i);   // v_readlane broadcast
        m = fmaxf(m, di + lt[i]);
      }
      delta = m + em;
    }
    // argmax over lanes (ties -> lowest index, matching jnp.argmax)
    float bv = delta; int bi = lane;
#pragma unroll
    for (int off = 16; off >= 1; off >>= 1) {
      float ov = __shfl_xor(bv, off, 32);
      int   oi = __shfl_xor(bi, off, 32);
      if (ov > bv || (ov == bv && oi < bi)) { bv = ov; bi = oi; }
    }
    int st = bi;
    // next state = argmax_j trans[st][j]
    float tv = s_trans[st][lane]; int ti = lane;
#pragma unroll
    for (int off = 16; off >= 1; off >>= 1) {
      float ov = __shfl_xor(tv, off, 32);
      int   oi = __shfl_xor(ti, off, 32);
      if (ov > tv || (ov == tv && oi < ti)) { tv = ov; ti = oi; }
    }
    if (lane == 0) s_state = ti;
  }
  __syncthreads();

  // ---- Phase D: AR dot + means[next_state] ------------------------------
  float partial = 0.0f;
  const int l0 = wave * 32;
  for (int l = l0; l < l0 + 32; l++)
    partial = fmaf(Eb[l * 32 + lane], Wm[l * 32 + lane], partial);
  s_red[wave][lane] = partial;
  __syncthreads();
  if (wave == 0) {
    float ar = 0.0f;
#pragma unroll
    for (int w = 0; w < 8; w++) ar += s_red[w][lane];
    out[b * 32 + lane] = ar + means[s_state * 32 + lane];
  }
}

extern "C" void kernel_launch(void* const* d_in, const int* in_sizes, int n_in,
                              void* d_out, int out_size, void* d_ws, size_t ws_size,
                              hipStream_t stream) {
  const float* E  = (const float*)d_in[0]; // emissions
  const float* TM = (const float*)d_in[1]; // transition_matrix
  const float* ID = (const float*)d_in[2]; // initial_distribution
  const float* MU = (const float*)d_in[3]; // means
  const float* CC = (const float*)d_in[4]; // covar_chol
  const float* W  = (const float*)d_in[5]; // W
  float* ws  = (float*)d_ws;
  float* out = (float*)d_out;

  hipLaunchKernelGGL(hmm_prep, dim3(1), dim3(32), 0, stream, TM, ID, MU, CC, ws);
  hipLaunchKernelGGL(hmm_main, dim3(256), dim3(256), 0, stream, E, W, MU, ws, out);

  (void)in_sizes; (void)n_in; (void)out_size; (void)ws_size;
}